// Patchifier_4518305595494
// MI455X (gfx1250) — compile-verified
//
#include <hip/hip_runtime.h>
#include <hip/hip_bf16.h>

// ---------------- problem constants ----------------
#define BN_TOT 16      // b*n = 2*8
#define IH 512
#define IW 512
#define GRID_H 32
#define GRID_W 32
#define CELL_H 16
#define CELL_W 16
#define NCELL 1024     // 32*32
#define PPF 256
#define C1 128
#define C2 64
#define FH 128
#define FW 128

typedef __attribute__((ext_vector_type(2))) float v2f;
typedef __attribute__((ext_vector_type(8))) float v8f;

// =====================================================================
// Kernel 1: Harris response per pixel + per-cell max/argmax.
// One block (256 threads) per 16x16 cell. LDS-tiled: 24x24 frame tile,
// 22x22 gradient-product tiles, then each thread does a 7x7 box sum.
// =====================================================================
__global__ void harris_cellmax(const float* __restrict__ frame,
                               float* __restrict__ cellval,
                               int* __restrict__ cellpos) {
  __shared__ float F[24][25];
  __shared__ float Pxx[22][23];
  __shared__ float Pyy[22][23];
  __shared__ float Pxy[22][23];
  __shared__ float rv[256];
  __shared__ int ri[256];

  const int blk = blockIdx.x;
  const int bn = blk >> 10;
  const int cell = blk & 1023;
  const int cy = cell >> 5, cx = cell & 31;
  const int y0 = cy * CELL_H, x0 = cx * CELL_W;
  const int tid = threadIdx.x;
  const float* img = frame + (size_t)bn * IH * IW;

  // Load 24x24 frame tile covering [y0-3, y0+20] x [x0-3, x0+20]; 0 outside.
  for (int idx = tid; idx < 24 * 24; idx += 256) {
    int ty = idx / 24, tx = idx % 24;
    int gy = y0 - 3 + ty, gx = x0 - 3 + tx;
    float v = 0.f;
    if (gy >= 0 && gy < IH && gx >= 0 && gx < IW) v = img[gy * IW + gx];
    F[ty][tx] = v;
  }
  __syncthreads();

  // Gradient products over [y0-3, y0+18] x [x0-3, x0+18] (22x22).
  // dx reflect only at xx==511 (dx = f(510)-f(509)); dy at yy==511.
  // Products outside the image are 0 (reduce_window zero padding).
  for (int idx = tid; idx < 22 * 22; idx += 256) {
    int py = idx / 22, px = idx % 22;
    int yy = y0 - 3 + py, xx = x0 - 3 + px;
    float pxx = 0.f, pyy = 0.f, pxy = 0.f;
    if (yy >= 0 && yy < IH && xx >= 0 && xx < IW) {
      float dx, dy;
      if (xx < IW - 1) dx = F[py][px + 1] - F[py][px];
      else             dx = F[py][513 - x0] - F[py][512 - x0];   // f(510)-f(509)
      if (yy < IH - 1) dy = F[py + 1][px] - F[py][px];
      else             dy = F[513 - y0][px] - F[512 - y0][px];   // f(510)-f(509)
      pxx = dx * dx; pyy = dy * dy; pxy = dx * dy;
    }
    Pxx[py][px] = pxx; Pyy[py][px] = pyy; Pxy[py][px] = pxy;
  }
  __syncthreads();

  // 7x7 box sums (count_include_pad semantics: always /49).
  const int ly = tid >> 4, lx = tid & 15;
  float sxx = 0.f, syy = 0.f, sxy = 0.f;
#pragma unroll
  for (int wy = 0; wy < 7; ++wy)
#pragma unroll
    for (int wx = 0; wx < 7; ++wx) {
      sxx += Pxx[ly + wy][lx + wx];
      syy += Pyy[ly + wy][lx + wx];
      sxy += Pxy[ly + wy][lx + wx];
    }
  const float inv49 = (float)(1.0 / 49.0);
  float Ixx = sxx * inv49, Iyy = syy * inv49, Ixy = sxy * inv49;
  float tr = Ixx + Iyy;
  float det = Ixx * Iyy - Ixy * Ixy;
  float g = det / (tr + 1e-8f);

  // Block max with first-index tie-break (matches jnp.argmax: flat = ly*16+lx = tid).
  rv[tid] = g; ri[tid] = tid;
  __syncthreads();
  for (int off = 128; off > 0; off >>= 1) {
    if (tid < off) {
      float v2 = rv[tid + off]; int i2 = ri[tid + off];
      float v1 = rv[tid];       int i1 = ri[tid];
      if (v2 > v1 || (v2 == v1 && i2 < i1)) { rv[tid] = v2; ri[tid] = i2; }
    }
    __syncthreads();
  }
  if (tid == 0) {
    int idx = ri[0];
    int yg = y0 + (idx >> 4);
    int xg = x0 + (idx & 15);
    cellval[bn * NCELL + cell] = rv[0];
    cellpos[bn * NCELL + cell] = (yg << 16) | xg;
  }
}

// =====================================================================
// Kernel 2: exact stable top-256 by ranking (1024 cells per image).
// rank_i = #{j : v_j > v_i or (v_j == v_i and j < i)} reproduces
// lax.top_k's stable descending order; ranks are a permutation.
// =====================================================================
__global__ void topk_coords(const float* __restrict__ cellval,
                            const int* __restrict__ cellpos,
                            float* __restrict__ coords) {
  __shared__ float vals[NCELL];
  const int bn = blockIdx.x;
  const int t = threadIdx.x;
  float v = cellval[bn * NCELL + t];
  vals[t] = v;
  __syncthreads();
  int rank = 0;
  for (int j = 0; j < NCELL; ++j) {
    float vj = vals[j];
    rank += (vj > v) || (vj == v && j < t);
  }
  if (rank < PPF) {
    int pos = cellpos[bn * NCELL + t];
    coords[(bn * PPF + rank) * 2 + 0] = (float)(pos >> 16);
    coords[(bn * PPF + rank) * 2 + 1] = (float)(pos & 0xFFFF);
  }
}

// =====================================================================
// Kernel 3: patches_f via WMMA. For one patch, all 9 bilinear samples'
// corners lie in a 4x4 pixel neighborhood (integer offsets => shared
// fractional weights), so patches_f[p] = W(9x16) x V(16x128).
// One wave per (bn, p, channel-tile of 16): 4 chained
// v_wmma_f32_16x16x4_f32 over the K=16 neighborhood.
// Axis-swap quirk (exact since FH==FW=128): col = y/4 + r[i], row = x/4 + r[j].
// =====================================================================
__global__ void patchf_wmma(const float* __restrict__ fmap,
                            const float* __restrict__ coords,
                            float* __restrict__ pf) {
  const int wave = blockIdx.x * (blockDim.x >> 5) + (threadIdx.x >> 5);
  const int lane = threadIdx.x & 31;
  const int ct = wave & 7;
  const int p = (wave >> 3) & 255;
  const int bn = wave >> 11;

  float y = coords[(bn * PPF + p) * 2 + 0];
  float x = coords[(bn * PPF + p) * 2 + 1];
  float cyf = y * 0.25f, cxf = x * 0.25f;
  float fyc = floorf(cyf);   // col base (from y)
  float fxr = floorf(cxf);   // row base (from x)
  float wc1 = cyf - fyc;     // col fraction
  float wr1 = cxf - fxr;     // row fraction
  int col_base = (int)fyc - 1;
  int row_base = (int)fxr - 1;

  const int hi = lane >> 4;
  const int nl = lane & 15;
  const int s = nl;               // A-matrix row (sample index, rows 9..15 zero)
  const int c = ct * 16 + nl;     // channel (B/D column)
  const int ii = s / 3, jj = s % 3;
  const bool srow = (s < 9);
  const float* fimg = fmap + ((size_t)bn * C1 + c) * (FH * FW);

  auto wrow = [&](int a) -> float {
    if (!srow) return 0.f;
    if (a == jj) return 1.f - wr1;
    if (a == jj + 1) return wr1;
    return 0.f;
  };
  auto wcol = [&](int b) -> float {
    if (!srow) return 0.f;
    if (b == ii) return 1.f - wc1;
    if (b == ii + 1) return wc1;
    return 0.f;
  };
  auto gather = [&](int a, int b) -> float {
    int r = row_base + a, q = col_base + b;
    bool ok = (r >= 0 && r < FH && q >= 0 && q < FW);
    int rc = min(max(r, 0), FH - 1);
    int qc = min(max(q, 0), FW - 1);
    float v = fimg[rc * FW + qc];
    return ok ? v : 0.f;
  };

#if defined(__has_builtin) && __has_builtin(__builtin_amdgcn_wmma_f32_16x16x4_f32)
  // A layout: lane L holds A[L%16][K = v + 2*(L>=16)]; B: B[K][L%16]; D: D[v+8*hi][L%16].
  v8f acc = {};
#pragma unroll
  for (int kk = 0; kk < 4; ++kk) {     // K-chunk = neighborhood row a = kk
    int b0 = 2 * hi, b1 = 2 * hi + 1;  // neighborhood cols for this lane half
    v2f av, bv;
    float wr_a = wrow(kk);
    av.x = wr_a * wcol(b0);
    av.y = wr_a * wcol(b1);
    bv.x = gather(kk, b0);
    bv.y = gather(kk, b1);
    acc = __builtin_amdgcn_wmma_f32_16x16x4_f32(
        /*neg_a=*/false, av, /*neg_b=*/false, bv,
        /*c_mod=*/(short)0, acc, /*reuse_a=*/false, /*reuse_b=*/false);
  }
#pragma unroll
  for (int v = 0; v < 8; ++v) {
    int M = v + 8 * hi;
    if (M < 9) {
      pf[((size_t)(bn * PPF + p) * C1 + c) * 9 + M] = acc[v];
    }
  }
#else
  // Scalar fallback (also used by the host-side semantic pass).
  if (hi == 0) {
    for (int sI = 0; sI < 9; ++sI) {
      int i2 = sI / 3, j2 = sI % 3;
      float wy0 = 1.f - wr1, wy1v = wr1;
      float wx0 = 1.f - wc1, wx1v = wc1;
      float val = gather(j2, i2)         * (wy0 * wx0)
                + gather(j2, i2 + 1)     * (wy0 * wx1v)
                + gather(j2 + 1, i2)     * (wy1v * wx0)
                + gather(j2 + 1, i2 + 1) * (wy1v * wx1v);
      pf[((size_t)(bn * PPF + p) * C1 + c) * 9 + sI] = val;
    }
  }
#endif
}

// =====================================================================
// Kernel 4: patches_c — single bilinear sample per (bn, p, channel).
// =====================================================================
__global__ void patchc_gather(const float* __restrict__ cmap,
                              const float* __restrict__ coords,
                              float* __restrict__ pc) {
  int tid = blockIdx.x * blockDim.x + threadIdx.x;
  if (tid >= BN_TOT * PPF * C2) return;
  int c = tid & 63;
  int p = (tid >> 6) & 255;
  int bn = tid >> 14;
  float y = coords[(bn * PPF + p) * 2 + 0];
  float x = coords[(bn * PPF + p) * 2 + 1];
  float ix = y * 0.25f;  // column (axis swap)
  float iy = x * 0.25f;  // row
  float x0f = floorf(ix), y0f = floorf(iy);
  float wx1 = ix - x0f, wy1 = iy - y0f;
  float wx0 = 1.f - wx1, wy0 = 1.f - wy1;
  int xi = (int)x0f, yi = (int)y0f;
  const float* img = cmap + ((size_t)bn * C2 + c) * (FH * FW);
  auto g = [&](int r, int q) -> float {
    bool ok = (r >= 0 && r < FH && q >= 0 && q < FW);
    int rc = min(max(r, 0), FH - 1);
    int qc = min(max(q, 0), FW - 1);
    float v = img[rc * FW + qc];
    return ok ? v : 0.f;
  };
  float val = g(yi, xi)         * (wy0 * wx0)
            + g(yi, xi + 1)     * (wy0 * wx1)
            + g(yi + 1, xi)     * (wy1 * wx0)
            + g(yi + 1, xi + 1) * (wy1 * wx1);
  pc[tid] = val;
}

// =====================================================================
// Kernel 5: fmap passthrough copy (dominant traffic: 134MB+134MB).
// float4 grid-stride + global_prefetch_b8.
// =====================================================================
__global__ void copy_fmap(const float4* __restrict__ src,
                          float4* __restrict__ dst, long n4) {
  long i = (long)blockIdx.x * blockDim.x + threadIdx.x;
  long stride = (long)gridDim.x * blockDim.x;
  for (; i < n4; i += stride) {
    if (i + stride < n4) __builtin_prefetch(&src[i + stride], 0, 1);
    dst[i] = src[i];
  }
}

// =====================================================================
extern "C" void kernel_launch(void* const* d_in, const int* in_sizes, int n_in,
                              void* d_out, int out_size, void* d_ws, size_t ws_size,
                              hipStream_t stream) {
  (void)in_sizes; (void)n_in; (void)out_size; (void)ws_size;
  const float* frame = (const float*)d_in[0];  // [16][512][512]
  const float* fmap  = (const float*)d_in[1];  // [16][128][128][128]
  const float* cmap  = (const float*)d_in[2];  // [16][64][128][128]

  float* out = (float*)d_out;
  float* coords = out;                                   // 16*256*2      = 8192
  float* pf = out + 8192;                                // 16*256*128*9  = 4718592
  float* pc = out + 8192 + 4718592;                      // 16*256*64     = 262144
  float* fout = out + 8192 + 4718592 + 262144;           // 16*128*128*128

  float* cellval = (float*)d_ws;                                   // 16384 floats
  int* cellpos = (int*)((char*)d_ws + (size_t)BN_TOT * NCELL * 4); // 16384 ints

  harris_cellmax<<<BN_TOT * NCELL, 256, 0, stream>>>(frame, cellval, cellpos);
  topk_coords<<<BN_TOT, NCELL, 0, stream>>>(cellval, cellpos, coords);
  // 16*256*8 waves, 8 waves (256 threads) per block
  patchf_wmma<<<BN_TOT * PPF, 256, 0, stream>>>(fmap, coords, pf);
  patchc_gather<<<(BN_TOT * PPF * C2 + 255) / 256, 256, 0, stream>>>(cmap, coords, pc);
  copy_fmap<<<4096, 256, 0, stream>>>((const float4*)fmap, (float4*)fout,
                                      (long)BN_TOT * C1 * FH * FW / 4);
}